// SpatialFusion_45921790329193
// MI455X (gfx1250) — compile-verified
//
#include <hip/hip_runtime.h>

// Problem constants (from reference)
#define B_  8
#define KA  4
#define C_  64
#define H_  100
#define W_  352

// Tile: 2 channels x 2 H-rows x full W per block, +1 halo each side for 3x3x3 SAME conv
#define TC  2
#define TH  2
#define SC  (TC + 2)        // 4 staged channels
#define SH  (TH + 2)        // 4 staged rows
#define SW  (W_ + 4)        // 356: 1+1 halo + 2 pad (alignment); only 0..353 used

typedef float v2f __attribute__((ext_vector_type(2)));
typedef float v8f __attribute__((ext_vector_type(8)));

__global__ __launch_bounds__(256)
void SpatialFusion_conv3d_wmma(const float* __restrict__ x,
                               const int*   __restrict__ record_len,
                               const float* __restrict__ wconv,   // [1,2,3,3,3] = 54
                               const float* __restrict__ bias,    // [1]
                               float*       __restrict__ out)     // [8,64,100,352]
{
    __shared__ float s_feat[2 * SC * SH * SW];   // [field(max/mean)][cc][hh][ww] ~45.6 KB
    __shared__ float s_w[56];                    // conv taps, padded with zeros
    __shared__ int   s_dl[56];                   // per-tap LDS offset delta

    const int tid = threadIdx.x;
    const int CT = C_ / TC, HT = H_ / TH;
    int id = blockIdx.x;
    const int b  = id / (CT * HT);
    int r = id % (CT * HT);
    const int c0 = (r / HT) * TC;
    const int h0 = (r % HT) * TH;

    // Ragged segment start/length (generic: prefix-sum of record_len)
    int start = 0;
    for (int j = 0; j < b; ++j) start += record_len[j];
    const int rl = record_len[b];
    const float inv_rl = 1.0f / (float)rl;

    // Stage conv weights + per-tap LDS offset deltas (taps 54,55 are zero-weight pads)
    if (tid < 56) {
        if (tid < 54) {
            int k  = tid;
            int ch = k / 27;            // 0 = max field, 1 = mean field
            int rr = k % 27;
            int dc = rr / 9 - 1;
            int dh = (rr % 9) / 3 - 1;
            int dw = rr % 3 - 1;
            s_w[tid]  = wconv[k];
            // field stride = SC*SH*SW = 16*SW; so delta = (ch*16 + dc*4 + dh)*SW + dw
            s_dl[tid] = (ch * 16 + dc * 4 + dh) * SW + dw;
        } else {
            s_w[tid]  = 0.0f;           // zero weight => pad taps contribute nothing
            s_dl[tid] = 0;              // safe in-range read
        }
    }

    // Stage max & mean tile with halo (zero outside bounds == SAME zero padding).
    const size_t plane = (size_t)C_ * H_ * W_;
    const float* xg = x + (size_t)start * plane;
    for (int idx = tid; idx < SC * SH * 354; idx += 256) {
        int cc = idx / (SH * 354);
        int r2 = idx % (SH * 354);
        int hh = r2 / 354;
        int ww = r2 % 354;
        int c = c0 - 1 + cc, h = h0 - 1 + hh, w = ww - 1;
        float mx = 0.0f, mn = 0.0f;
        if ((unsigned)c < (unsigned)C_ && (unsigned)h < (unsigned)H_ &&
            (unsigned)w < (unsigned)W_) {
            size_t off = ((size_t)c * H_ + h) * W_ + w;
            float v0 = xg[off];
            __builtin_prefetch(xg + off + 2048, 0, 0);   // global_prefetch_b8 (streaming hint)
            float vmax = v0, vsum = v0;
            #pragma unroll
            for (int k = 1; k < KA; ++k) {
                if (k < rl) {
                    float v = xg[(size_t)k * plane + off];
                    vmax = fmaxf(vmax, v);
                    vsum += v;
                }
            }
            mx = vmax;
            mn = vsum * inv_rl;
        }
        s_feat[(cc * SH + hh) * SW + ww]                  = mx;  // field 0: max
        s_feat[(SC * SH * SW) + (cc * SH + hh) * SW + ww] = mn;  // field 1: mean
    }
    __syncthreads();

    const float bval = bias[0];
    const int wave = tid >> 5;
    const int lane = tid & 31;
    const int n  = lane & 15;   // output W position within segment (B-matrix column N)
    const int kp = lane >> 4;   // 0 -> taps K0,K1 ; 1 -> taps K2,K3 (B-matrix row layout)

    // Transposed im2col mapping:
    //   A[m,k]  = w[k]  broadcast over M (weights, lane-invariant operands)
    //   B[k,n]  = feat tap k at W position wbase+n (per-lane LDS gather)
    //   D[m,n]  = out[wbase+n] for every m -> VGPR0 lanes 0-15 = coalesced output row.
    // 4 rows * 22 segments = 88 iterations / 8 waves; uniform per-wave control flow
    // keeps EXEC all-ones through the WMMA chain (ISA requirement).
    for (int it = wave; it < TC * TH * (W_ / 16); it += 8) {
        int row   = it / (W_ / 16);      // 0..3
        int wseg  = it % (W_ / 16);      // 0..21
        int cl = row >> 1, hl = row & 1;
        int wbase = wseg * 16;
        int base = ((cl + 1) * SH + (hl + 1)) * SW + 1 + wbase + n;

        v8f acc = {0.f, 0.f, 0.f, 0.f, 0.f, 0.f, 0.f, 0.f};
        #pragma unroll
        for (int ks = 0; ks < 14; ++ks) {          // 56 taps / K=4 per WMMA
            int k0 = ks * 4 + kp * 2;
            v2f a, bb;
            a.x  = s_w[k0];                        // A[m][2kp]   = weight, all m
            a.y  = s_w[k0 + 1];                    // A[m][2kp+1]
            bb.x = s_feat[base + s_dl[k0]];        // B[2kp][n]   = im2col tap
            bb.y = s_feat[base + s_dl[k0 + 1]];    // B[2kp+1][n]
            // D = A(16x4) x B(4x16) + C  -> v_wmma_f32_16x16x4_f32
            acc = __builtin_amdgcn_wmma_f32_16x16x4_f32(
                      false, a, false, bb, (short)0, acc, false, false);
        }

        // D row M=0 lives in VGPR0 of lanes 0-15 (N=lane): one coalesced 64B store.
        if (lane < 16) {
            size_t o = (((size_t)b * C_ + (c0 + cl)) * H_ + (h0 + hl)) * W_ + wbase + n;
            out[o] = fmaxf(acc[0] + bval, 0.0f);   // bias + ReLU
        }
    }
}

extern "C" void kernel_launch(void* const* d_in, const int* in_sizes, int n_in,
                              void* d_out, int out_size, void* d_ws, size_t ws_size,
                              hipStream_t stream) {
    const float* x    = (const float*)d_in[0];   // [32,64,100,352] f32
    const int*   rlen = (const int*)  d_in[1];   // [8] i32
    const float* w    = (const float*)d_in[2];   // [1,2,3,3,3] f32
    const float* bias = (const float*)d_in[3];   // [1] f32
    float*       out  = (float*)d_out;           // [8,64,100,352] f32

    const int blocks = B_ * (C_ / TC) * (H_ / TH);   // 8*32*50 = 12800
    hipLaunchKernelGGL(SpatialFusion_conv3d_wmma, dim3(blocks), dim3(256), 0, stream,
                       x, rlen, w, bias, out);
}